// NXROAttentiveModel_49855980372440
// MI455X (gfx1250) — compile-verified
//
#include <hip/hip_runtime.h>
#include <math.h>

typedef __attribute__((ext_vector_type(2))) float v2f;
typedef __attribute__((ext_vector_type(8))) float v8f;

#define NV    10   // n_vars
#define KDIM  50   // 5 basis * 10 vars (padded to 52 = 13*4 for WMMA K)
#define NSTEP 13   // ceil(50/4)

__global__ __launch_bounds__(256)
void nxro_fused_kernel(const float* __restrict__ x,
                       const float* __restrict__ t_years,
                       const float* __restrict__ Lb,
                       const float* __restrict__ wq,
                       const float* __restrict__ wk,
                       const float* __restrict__ wv,
                       const float* __restrict__ wo,
                       const float* __restrict__ alpha_w,
                       float* __restrict__ out,
                       int B, int iters, int totalWaves)
{
    __shared__ float sm[8][16][12];          // per-wave softmax tile (padded)

    const int lane = threadIdx.x & 31;
    const int wib  = threadIdx.x >> 5;       // wave in block (8 waves)
    const int hi   = lane >> 4;              // half-wave: 0 or 1
    const int n    = lane & 15;              // A-row / B,D-column index
    const int waveId = blockIdx.x * (blockDim.x >> 5) + wib;

    // --- uniform scalar constants (scalar-load friendly) ---
    float c1 = 0.f, c2 = 0.f;
    #pragma unroll
    for (int i = 0; i < 32; ++i) { c1 += wq[i] * wk[i]; c2 += wv[i] * wo[i]; }
    c1 *= 0.17677669529663687f;              // 1/sqrt(32)
    float aw0 = alpha_w[0], aw1 = alpha_w[1], aw2 = alpha_w[2],
          aw3 = alpha_w[3], aw4 = alpha_w[4];

    // --- preload constant B fragments: W[kk][u] = Lb[kk/10][u][kk%10] ---
    // f32 4x16 B layout: VGPR.x holds K=4s (lanes 0-15) / K=4s+2 (lanes 16-31)
    //                    VGPR.y holds K=4s+1 / K=4s+3
    v2f bf[NSTEP];
    #pragma unroll
    for (int s = 0; s < NSTEP; ++s) {
        float b0l = 0.f, b1l = 0.f, b0h = 0.f, b1h = 0.f;
        if (n < NV) {
            b0l = Lb[((4*s  ) / NV) * NV * NV + n * NV + ((4*s  ) % NV)];
            b1l = Lb[((4*s+1) / NV) * NV * NV + n * NV + ((4*s+1) % NV)];
            if (4*s+2 < KDIM) b0h = Lb[((4*s+2) / NV) * NV * NV + n * NV + ((4*s+2) % NV)];
            if (4*s+3 < KDIM) b1h = Lb[((4*s+3) / NV) * NV * NV + n * NV + ((4*s+3) % NV)];
        }
        bf[s].x = hi ? b0h : b0l;
        bf[s].y = hi ? b1h : b1l;
    }

    const float TWO_PI = 6.283185307179586f;

    for (int it = 0; it < iters; ++it) {
        int  tile = waveId + it * totalWaves;
        int  tb   = tile * 16;
        bool tileValid = tb < B;

        // each lane owns batch row tb + (lane&15); both halves duplicate rows
        int  row = tb + n;
        bool rowValid = tileValid && (row < B);

        float xr[NV];
        #pragma unroll
        for (int j = 0; j < NV; ++j) xr[j] = rowValid ? x[row * NV + j] : 0.f;
        float tv = rowValid ? t_years[row] : 0.f;

        float w  = TWO_PI * tv;
        float w2 = w + w;
        float emb[5];
        emb[0] = 1.f;
        emb[1] = __cosf(w);  emb[2] = __sinf(w);
        emb[3] = __cosf(w2); emb[4] = __sinf(w2);

        // --- dxdt tile via 13x V_WMMA_F32_16X16X4_F32 (exact f32) ---
        v8f acc = {};
        #pragma unroll
        for (int s = 0; s < NSTEP; ++s) {
            // A 16x4 f32 layout: VGPR.x = K=4s (lanes<16) / 4s+2 (lanes>=16)
            float ax0 = emb[(4*s  ) / NV] * xr[(4*s  ) % NV];
            float ay0 = emb[(4*s+1) / NV] * xr[(4*s+1) % NV];
            float ax1 = (4*s+2 < KDIM) ? emb[(4*s+2) / NV] * xr[(4*s+2) % NV] : 0.f;
            float ay1 = (4*s+3 < KDIM) ? emb[(4*s+3) / NV] * xr[(4*s+3) % NV] : 0.f;
            v2f a;
            a.x = hi ? ax1 : ax0;
            a.y = hi ? ay1 : ay0;
            acc = __builtin_amdgcn_wmma_f32_16x16x4_f32(
                      false, a, false, bf[s], (short)0, acc, false, false);
        }

        // --- collapsed attention: per row r, per v: softmax over u in {0,1,v} ---
        float z = aw0 + aw1*emb[1] + aw2*emb[2] + aw3*emb[3] + aw4*emb[4];
        float alpha = 1.f / (1.f + __expf(-z));
        float g = alpha * c2;

        float s0 = c1 * xr[0];
        float s1 = c1 * xr[1];
        float smrow[NV];
        #pragma unroll
        for (int v = 0; v < NV; ++v) {
            float a0 = s0 * xr[v];
            float a1 = s1 * xr[v];
            if (v >= 2) {
                float av = c1 * xr[v] * xr[v];
                float m  = fmaxf(fmaxf(a0, a1), av);
                float e0 = __expf(a0 - m), e1 = __expf(a1 - m), ev = __expf(av - m);
                smrow[v] = g * (e0*xr[0] + e1*xr[1] + ev*xr[v]) / (e0 + e1 + ev);
            } else {
                float m  = fmaxf(a0, a1);
                float e0 = __expf(a0 - m), e1 = __expf(a1 - m);
                smrow[v] = g * (e0*xr[0] + e1*xr[1]) / (e0 + e1);
            }
        }
        if (hi == 0) {
            #pragma unroll
            for (int v = 0; v < NV; ++v) sm[wib][n][v] = smrow[v];
        }
        __syncthreads();

        // --- combine: D lane(16h+n) holds rows 8h+j, col n in acc[j] ---
        if (tileValid && n < NV) {
            #pragma unroll
            for (int j = 0; j < 8; ++j) {
                int rr   = 8 * hi + j;
                int orow = tb + rr;
                if (orow < B) out[orow * NV + n] = acc[j] + sm[wib][rr][n];
            }
        }
        __syncthreads();
    }
}

extern "C" void kernel_launch(void* const* d_in, const int* in_sizes, int n_in,
                              void* d_out, int out_size, void* d_ws, size_t ws_size,
                              hipStream_t stream) {
    const float* x  = (const float*)d_in[0];
    const float* t  = (const float*)d_in[1];
    const float* Lb = (const float*)d_in[2];
    const float* wq = (const float*)d_in[3];
    const float* wk = (const float*)d_in[4];
    const float* wv = (const float*)d_in[5];
    const float* wo = (const float*)d_in[6];
    const float* aw = (const float*)d_in[7];
    float* out = (float*)d_out;

    int B = in_sizes[1];                     // t_years has B elements
    int tiles = (B + 15) / 16;
    const int wavesPerBlock = 8;             // 256 threads, wave32
    int blocks = 1024;
    if (tiles < blocks * wavesPerBlock)
        blocks = (tiles + wavesPerBlock - 1) / wavesPerBlock;
    int totalWaves = blocks * wavesPerBlock;
    int iters = (tiles + totalWaves - 1) / totalWaves;

    nxro_fused_kernel<<<blocks, 256, 0, stream>>>(
        x, t, Lb, wq, wk, wv, wo, aw, out, B, iters, totalWaves);
}